// AutoregressiveDecoder_24799141167144
// MI455X (gfx1250) — compile-verified
//
#include <hip/hip_runtime.h>
#include <math.h>

// ---------------- problem constants (fixed shapes from reference) ----------
constexpr int Bc  = 4;
constexpr int Nc  = 256;
constexpr int Hc  = 192;
constexpr int H3c = 3 * Hc;   // 576
constexpr int Fc  = 64;
constexpr float EPSc = 1e-6f;

constexpr int L_SIZE  = Bc * Nc * Nc;          // 262144
constexpr int NF_OFF  = L_SIZE;                // node_features start
constexpr int NF_SIZE = Bc * Nc * Fc;          // 65536
constexpr int LL_OFF  = NF_OFF + NF_SIZE;      // 327680

constexpr int BS     = 512;                    // scan block: 16 wave32s
constexpr int NWAVES = BS / 32;

// LDS budget for the scan kernel (one batch per workgroup, CDNA5: 320KB/WGP)
constexpr int SMEM_FLOATS =
    Nc * Hc      // C = S @ W1b incremental cache (192 KB)
  + Hc           // Ssum
  + Hc           // z_s
  + H3c          // ghz = z@W_hh+b_hh (fixed)
  + Hc           // zc  = z@W1c+b1   (fixed)
  + Hc           // m
  + H3c          // gx1
  + Hc           // h_prov
  + Hc           // base
  + 2 * Nc       // tgt row double buffer (TDM prefetch target)
  + Hc           // msg
  + H3c          // gx2
  + H3c          // ghp
  + Hc           // h_new
  + Hc           // W2 copy
  + NWAVES       // per-wave ll
  + 4;           // misc: ll_acc, b2, ...
constexpr int SMEM_WORDS = SMEM_FLOATS + (Nc + 1) + (Nc * 8 + 8); // plist+npar, Abits+ancb
constexpr size_t SMEM_BYTES = (size_t)SMEM_WORDS * 4;

__device__ __forceinline__ float sigm(float x) { return 1.0f / (1.0f + __expf(-x)); }

// ---------------------------------------------------------------------------
// Tensor Data Mover: async DMA of one 256-float target row -> LDS.
// D# packed per CDNA5 ISA (group0/group1 bitfields).
// This toolchain exposes the 6-arg builtin:
//   (uint32x4 g0, int32x8 g1, int32x4 g2, int32x4 g3, int32x8 pad, i32 cpol)
// ---------------------------------------------------------------------------
typedef unsigned int u32x4 __attribute__((ext_vector_type(4)));
typedef int          i32x4 __attribute__((ext_vector_type(4)));
typedef int          i32x8 __attribute__((ext_vector_type(8)));

#if __has_builtin(__builtin_amdgcn_tensor_load_to_lds) && __has_builtin(__builtin_amdgcn_s_wait_tensorcnt)
#define HAVE_TDM 1
#else
#define HAVE_TDM 0
#endif

#if HAVE_TDM
__device__ __forceinline__ void tdm_load_tgt_row(const float* gsrc, unsigned lds_byte_off) {
  unsigned long long ga = (unsigned long long)(uintptr_t)gsrc;
  u32x4 g0;
  g0[0] = 1u;                                            // count=1 valid user descriptor
  g0[1] = lds_byte_off;                                  // lds_addr (bytes)
  g0[2] = (unsigned)(ga & 0xffffffffu);                  // global_addr[31:0]
  g0[3] = (unsigned)((ga >> 32) & 0x01ffffffu)           // global_addr[56:32]
        | 0x80000000u;                                   // type=2 ("image")
  i32x8 g1;
  g1[0] = 0x00020000;                                    // data_size=4B; no multicast/flags
  g1[1] = (int)((unsigned)Nc << 16);                     // tensor_dim0 = 256 (lo16)
  g1[2] = 0x00010000;                                    // tensor_dim0 hi=0 | tensor_dim1 = 1
  g1[3] = (int)((unsigned)Nc << 16);                     // tensor_dim1 hi=0 | tile_dim0 = 256
  g1[4] = 1;                                             // tile_dim1 = 1, tile_dim2 = 0
  g1[5] = Nc;                                            // tensor_dim0_stride = 256 (lo32)
  g1[6] = (int)((unsigned)Nc << 16);                     // stride0 hi=0 | tensor_dim1_stride lo
  g1[7] = 0;
  i32x4 gz4 = {0, 0, 0, 0};                              // groups 2/3 unused (2D tensor)
  i32x8 gz8 = {0, 0, 0, 0, 0, 0, 0, 0};
  __builtin_amdgcn_tensor_load_to_lds(g0, g1, gz4, gz4, gz8, 0);
}
#endif

// ---------------------------------------------------------------------------
// Kernel 1: copy L = targets, zero the ll accumulator slot
// ---------------------------------------------------------------------------
__global__ void init_out_kernel(const float* __restrict__ tgt, float* __restrict__ out) {
  int i = blockIdx.x * blockDim.x + threadIdx.x;
  if (i < L_SIZE) out[i] = tgt[i];
  if (i == 0)     out[LL_OFF] = 0.0f;
}

// ---------------------------------------------------------------------------
// Kernel 2: full autoregressive scan; one workgroup per batch element.
// Serial over t, parallel over H / 3H / candidate-parents inside each step.
// TDM double-buffers the next step's target row so the only dependent global
// access on the 255-step critical path is overlapped with compute.
// ---------------------------------------------------------------------------
__global__ __launch_bounds__(BS, 1)
void scan_kernel(const float* __restrict__ z, const float* __restrict__ tgt,
                 const float* __restrict__ W_init, const float* __restrict__ b_init,
                 const float* __restrict__ W_ih,   const float* __restrict__ W_hh,
                 const float* __restrict__ b_ih,   const float* __restrict__ b_hh,
                 const float* __restrict__ W1,     const float* __restrict__ b1,
                 const float* __restrict__ W2,     const float* __restrict__ b2,
                 float* __restrict__ S_g, float* __restrict__ out_ll) {
  extern __shared__ char smem_raw[];
  float* Cc     = (float*)smem_raw;          // [Nc*Hc]
  float* Ssum   = Cc     + Nc * Hc;          // [Hc]
  float* zs     = Ssum   + Hc;               // [Hc]
  float* ghz    = zs     + Hc;               // [H3c]
  float* zc     = ghz    + H3c;              // [Hc]
  float* ms     = zc     + Hc;               // [Hc]
  float* gx1    = ms     + Hc;               // [H3c]
  float* hprov  = gx1    + H3c;              // [Hc]
  float* bases  = hprov  + Hc;               // [Hc]
  float* tgtbuf = bases  + Hc;               // [2*Nc] double buffer
  float* msgs   = tgtbuf + 2 * Nc;           // [Hc]
  float* gx2    = msgs   + Hc;               // [H3c]
  float* ghp    = gx2    + H3c;              // [H3c]
  float* hnew   = ghp    + H3c;              // [Hc]
  float* W2s    = hnew   + Hc;               // [Hc]
  float* wll    = W2s    + Hc;               // [NWAVES]
  float* misc   = wll    + NWAVES;           // [4]  misc[0]=ll, misc[1]=b2
  int*      plist = (int*)(misc + 4);        // [Nc]
  int*      npar  = plist + Nc;              // [1]
  unsigned* Abits = (unsigned*)(npar + 1);   // [Nc*8] ancestor bitsets
  unsigned* ancb  = Abits + Nc * 8;          // [8]

  const int b    = blockIdx.x;
  const int tid  = threadIdx.x;
  const int lane = tid & 31;
  const int wave = tid >> 5;

  const unsigned tgt_lds_base = (unsigned)((char*)tgtbuf - (char*)smem_raw);

  // Warm L2 with the streamed weight matrices (global_prefetch_b8)
  {
    const size_t wbytes = (size_t)Hc * H3c * 4;           // W_ih / W_hh
    for (size_t off = (size_t)tid * 128; off < wbytes; off += (size_t)BS * 128) {
      __builtin_prefetch((const char*)W_ih + off, 0, 1);
      __builtin_prefetch((const char*)W_hh + off, 0, 1);
      __builtin_prefetch((const char*)W1 + off, 0, 1);    // W1 is (3H,H): same byte count
    }
  }

#if HAVE_TDM
  // Kick off async DMA of target row t=1 while we do the t=0 precompute.
  if (wave == 0)
    tdm_load_tgt_row(tgt + ((size_t)b * Nc + 1) * Nc, tgt_lds_base + (1 & 1) * Nc * 4);
#endif

  // ---------------- per-batch precompute (t = 0) ----------------
  for (int i = tid; i < Hc; i += BS) { zs[i] = z[b * Hc + i]; W2s[i] = W2[i]; }
  for (int i = tid; i < Nc * 8; i += BS) Abits[i] = 0u;
  if (tid == 0) { misc[0] = 0.0f; misc[1] = b2[0]; }
  __syncthreads();

  for (int c = tid; c < H3c; c += BS) {          // ghz = z @ W_hh + b_hh (loop-invariant)
    float acc = b_hh[c];
    #pragma unroll 8
    for (int k = 0; k < Hc; ++k) acc += zs[k] * W_hh[k * H3c + c];
    ghz[c] = acc;
  }
  for (int i = tid; i < Hc; i += BS) {           // zc = z@W1c + b1 ; h0 = tanh(z@W_init+b_init)
    float a1 = b1[i], a0 = b_init[i];
    #pragma unroll 8
    for (int k = 0; k < Hc; ++k) {
      a1 += zs[k] * W1[(2 * Hc + k) * Hc + i];
      a0 += zs[k] * W_init[k * Hc + i];
    }
    zc[i] = a1;
    float h0 = tanhf(a0);
    hnew[i] = h0;
    Ssum[i] = h0;
    S_g[((size_t)b * Nc + 0) * Hc + i] = h0;
  }
  __syncthreads();
  for (int i = tid; i < Hc; i += BS) {           // C[0] = h0 @ W1b
    float acc = 0.0f;
    #pragma unroll 8
    for (int k = 0; k < Hc; ++k) acc += hnew[k] * W1[(Hc + k) * Hc + i];
    Cc[i] = acc;
  }
  __threadfence();
  __syncthreads();

  // ---------------- serial scan t = 1..Nc-1 ----------------
  for (int t = 1; t < Nc; ++t) {
    float* tgts = tgtbuf + (t & 1) * Nc;         // row t (DMA'd a step ahead)

    if (tid < 8)  ancb[tid] = 0u;
    if (tid == 0) npar[0] = 0;
#if HAVE_TDM
    if (wave == 0) __builtin_amdgcn_s_wait_tensorcnt(0);   // retire row-t DMA
#else
    for (int j = tid; j < Nc; j += BS) tgts[j] = tgt[((size_t)b * Nc + t) * Nc + j];
#endif
    if (tid < Hc) ms[tid] = Ssum[tid] * (1.0f / (float)t); // mean over t committed nodes
    __syncthreads();

#if HAVE_TDM
    // Launch DMA for row t+1 into the other buffer; overlaps this whole step.
    if (wave == 0 && (t + 1) < Nc)
      tdm_load_tgt_row(tgt + ((size_t)b * Nc + (t + 1)) * Nc,
                       tgt_lds_base + ((t + 1) & 1) * Nc * 4);
#endif

    // ancestor bits of parents (reach>0) + parent list
    for (int j = tid; j < Nc; j += BS) {
      if (tgts[j] > 0.5f) {
        int pos = atomicAdd(npar, 1);
        plist[pos] = j;
        #pragma unroll
        for (int w = 0; w < 8; ++w) atomicOr(&ancb[w], Abits[j * 8 + w]);
      }
    }
    // gx1 = m @ W_ih + b_ih
    for (int c = tid; c < H3c; c += BS) {
      float acc = b_ih[c];
      #pragma unroll 8
      for (int k = 0; k < Hc; ++k) acc += ms[k] * W_ih[k * H3c + c];
      gx1[c] = acc;
    }
    __syncthreads();

    // h_prov = GRU(m, z)   (gh part precomputed in ghz)
    if (tid < Hc) {
      float r  = sigm(gx1[tid] + ghz[tid]);
      float u  = sigm(gx1[Hc + tid] + ghz[Hc + tid]);
      float nn = tanhf(gx1[2 * Hc + tid] + r * ghz[2 * Hc + tid]);
      hprov[tid] = (1.0f - u) * nn + u * zs[tid];
    }
    __syncthreads();

    // base = h_prov @ W1a + zc  (broadcast part of the MLP hidden layer)
    for (int i = tid; i < Hc; i += BS) {
      float acc = zc[i];
      #pragma unroll 8
      for (int k = 0; k < Hc; ++k) acc += hprov[k] * W1[k * Hc + i];
      bases[i] = acc;
    }
    __syncthreads();

    // logits + log-likelihood over existing candidates j < t (waves over j)
    {
      float llw = 0.0f;
      for (int j = wave; j < t; j += NWAVES) {
        float acc = 0.0f;
        #pragma unroll
        for (int i = lane; i < Hc; i += 32) {
          float hd = Cc[j * Hc + i] + bases[i];
          hd = hd > 0.0f ? hd : 0.0f;
          acc += hd * W2s[i];
        }
        #pragma unroll
        for (int off = 16; off > 0; off >>= 1) acc += __shfl_down(acc, off, 32);
        if (lane == 0) {
          float p = sigm(acc + misc[1]);
          if ((ancb[j >> 5] >> (j & 31)) & 1u) p = 0.0f;   // SUP = 1.0
          p = fminf(fmaxf(p, EPSc), 1.0f - EPSc);
          float tg = tgts[j];
          llw += tg * logf(p) + (1.0f - tg) * log1pf(-p);
        }
      }
      if (lane == 0) wll[wave] = llw;
    }
    __syncthreads();
    if (tid == 0) {
      float s = 0.0f;
      #pragma unroll
      for (int w = 0; w < NWAVES; ++w) s += wll[w];
      misc[0] += s;
    }

    // msg = mean over parent states (sparse gather from global S)
    int np = npar[0];
    float invdeg = 1.0f / fmaxf((float)np, 1.0f);
    if (tid < Hc) {
      float acc = 0.0f;
      for (int q = 0; q < np; ++q)
        acc += S_g[((size_t)b * Nc + plist[q]) * Hc + tid];
      msgs[tid] = acc * invdeg;
    }
    __syncthreads();

    // gx2 = msg@W_ih + b_ih ; ghp = h_prov@W_hh + b_hh
    for (int c = tid; c < 2 * H3c; c += BS) {
      if (c < H3c) {
        float acc = b_ih[c];
        #pragma unroll 8
        for (int k = 0; k < Hc; ++k) acc += msgs[k] * W_ih[k * H3c + c];
        gx2[c] = acc;
      } else {
        int cc = c - H3c;
        float acc = b_hh[cc];
        #pragma unroll 8
        for (int k = 0; k < Hc; ++k) acc += hprov[k] * W_hh[k * H3c + cc];
        ghp[cc] = acc;
      }
    }
    __syncthreads();

    // h_new = GRU(msg, h_prov)
    if (tid < Hc) {
      float r  = sigm(gx2[tid] + ghp[tid]);
      float u  = sigm(gx2[Hc + tid] + ghp[Hc + tid]);
      float nn = tanhf(gx2[2 * Hc + tid] + r * ghp[2 * Hc + tid]);
      hnew[tid] = (1.0f - u) * nn + u * hprov[tid];
    }
    __syncthreads();

    // commit: S row t, running sum, incremental C row t = h_new @ W1b, A row t
    for (int i = tid; i < Hc; i += BS) {
      float hv = hnew[i];
      S_g[((size_t)b * Nc + t) * Hc + i] = hv;
      Ssum[i] += hv;
      float acc = 0.0f;
      #pragma unroll 8
      for (int k = 0; k < Hc; ++k) acc += hnew[k] * W1[(Hc + k) * Hc + i];
      Cc[t * Hc + i] = acc;
    }
    if (tid < 8) {                                  // A[t] = tgt OR anc (bitset)
      unsigned tb = 0u;
      #pragma unroll
      for (int bit = 0; bit < 32; ++bit)
        if (tgts[(tid << 5) + bit] > 0.5f) tb |= (1u << bit);
      Abits[t * 8 + tid] = tb | ancb[tid];
    }
    __threadfence();
    __syncthreads();
  }

  if (tid == 0) atomicAdd(out_ll, misc[0]);
}

// ---------------------------------------------------------------------------
// Kernel 3: node_features = S @ Wf + bf  via fp32 WMMA 16x16x4 (exact f32 MAC)
//   (B*N=1024, H=192) @ (H=192, F=64): 64 m-tiles x 4 n-tiles, 48 WMMAs/tile.
// ---------------------------------------------------------------------------
typedef __attribute__((ext_vector_type(2))) float v2f;
typedef __attribute__((ext_vector_type(8))) float v8f;

__global__ __launch_bounds__(128)
void feat_kernel(const float* __restrict__ S, const float* __restrict__ Wf,
                 const float* __restrict__ bf, float* __restrict__ out) {
  const int wave = threadIdx.x >> 5;
  const int lane = threadIdx.x & 31;
  const int tile = blockIdx.x * 4 + wave;   // 0..255
  const int m0   = (tile >> 2) * 16;        // row tile base in [0,1024)
  const int f0   = (tile & 3) * 16;         // col tile base in [0,64)
  const int half = lane >> 4;               // 0: K pair {0,1}; 1: K pair {2,3}
  const int l15  = lane & 15;

  v8f acc = {};
  for (int k0 = 0; k0 < Hc; k0 += 4) {
    const int ka = k0 + half * 2;
    v2f a, bb;
    // A (16x4 f32): lanes 0-15 -> K=0/1 in vgpr0/1; lanes 16-31 -> K=2/3 (ISA 7.12.2)
    a.x  = S[(size_t)(m0 + l15) * Hc + ka];
    a.y  = S[(size_t)(m0 + l15) * Hc + ka + 1];
    // B (4x16 f32): row-striped across lanes, symmetric to A
    bb.x = Wf[(size_t)ka * Fc + f0 + l15];
    bb.y = Wf[(size_t)(ka + 1) * Fc + f0 + l15];
    acc = __builtin_amdgcn_wmma_f32_16x16x4_f32(
        /*neg_a=*/false, a, /*neg_b=*/false, bb,
        /*c_mod=*/(short)0, acc, /*reuse_a=*/false, /*reuse_b=*/false);
  }
  // C/D layout: VGPR r, lanes 0-15 -> M=r; lanes 16-31 -> M=r+8
  const int col = f0 + l15;
  const float bias = bf[col];
  #pragma unroll
  for (int r = 0; r < 8; ++r) {
    const int row = m0 + r + half * 8;
    out[NF_OFF + (size_t)row * Fc + col] = acc[r] + bias;
  }
}

// ---------------------------------------------------------------------------
extern "C" void kernel_launch(void* const* d_in, const int* in_sizes, int n_in,
                              void* d_out, int out_size, void* d_ws, size_t ws_size,
                              hipStream_t stream) {
  (void)in_sizes; (void)n_in; (void)out_size; (void)ws_size;
  const float* z      = (const float*)d_in[0];
  const float* tgt    = (const float*)d_in[1];
  const float* W_init = (const float*)d_in[2];
  const float* b_init = (const float*)d_in[3];
  const float* W_ih   = (const float*)d_in[4];
  const float* W_hh   = (const float*)d_in[5];
  const float* b_ih   = (const float*)d_in[6];
  const float* b_hh   = (const float*)d_in[7];
  const float* W1     = (const float*)d_in[8];
  const float* b1     = (const float*)d_in[9];
  const float* W2     = (const float*)d_in[10];
  const float* b2     = (const float*)d_in[11];
  const float* Wf     = (const float*)d_in[12];
  const float* bf     = (const float*)d_in[13];

  float* out = (float*)d_out;
  float* S_g = (float*)d_ws;   // B*N*H f32 node-state workspace (768 KB)

  init_out_kernel<<<(L_SIZE + 255) / 256, 256, 0, stream>>>(tgt, out);
  scan_kernel<<<Bc, BS, SMEM_BYTES, stream>>>(z, tgt, W_init, b_init, W_ih, W_hh,
                                              b_ih, b_hh, W1, b1, W2, b2,
                                              S_g, out + LL_OFF);
  feat_kernel<<<64, 128, 0, stream>>>(S_g, Wf, bf, out);
}